// PaiNNInteraction_39170101739761
// MI455X (gfx1250) — compile-verified
//
#include <hip/hip_runtime.h>
#include <math.h>

#define NUM_NODES 20000
#define NUM_EDGES 640000
#define NODE_SIZE 128
#define EDGE_SIZE 20
#define OUT3      (3 * NODE_SIZE)   // 384
#define CUTOFF_R  5.0f
#define PI_F      3.14159265358979323846f

typedef __attribute__((ext_vector_type(2))) float v2f;
typedef __attribute__((ext_vector_type(8))) float v8f;

// D = A(16x4 f32) x B(4x16 f32) + C, f32 accumulate (CDNA5 V_WMMA_F32_16X16X4_F32)
__device__ __forceinline__ v8f wmma_f32_4(v2f a, v2f b, v8f c) {
    return __builtin_amdgcn_wmma_f32_16x16x4_f32(false, a, false, b, (short)0, c,
                                                 false, false);
}

// ---------------------------------------------------------------------------
// Kernel 1: scalar_output = silu(ns @ W1 + b1) @ W2 + b2   (20000 x 384)
// One block = 16 nodes, 256 threads = 8 waves.
// ---------------------------------------------------------------------------
__global__ void __launch_bounds__(256)
node_mlp_kernel(const float* __restrict__ ns,  // (N,128)
                const float* __restrict__ W1,  // (128,128)
                const float* __restrict__ b1,  // (128)
                const float* __restrict__ W2,  // (128,384)
                const float* __restrict__ b2,  // (384)
                float* __restrict__ so)        // (N,384) out
{
    __shared__ float Hs[16 * NODE_SIZE];       // silu hidden tile, 8 KB

    const int m0   = blockIdx.x * 16;
    const int wave = threadIdx.x >> 5;         // 0..7
    const int lane = threadIdx.x & 31;         // wave32
    const int half = lane >> 4;                // 0/1
    const int lrow = lane & 15;

    // -------- stage 1: hidden = silu(ns @ W1 + b1); wave w -> cols 16w..16w+15
    {
        const int n0 = wave * 16;
        v8f c = {};
#pragma unroll
        for (int k0 = 0; k0 < NODE_SIZE; k0 += 4) {
            const int k = k0 + 2 * half;
            v2f a, b;
            a.x = ns[(m0 + lrow) * NODE_SIZE + k + 0];
            a.y = ns[(m0 + lrow) * NODE_SIZE + k + 1];
            b.x = W1[(k + 0) * NODE_SIZE + n0 + lrow];
            b.y = W1[(k + 1) * NODE_SIZE + n0 + lrow];
            c = wmma_f32_4(a, b, c);
        }
#pragma unroll
        for (int v = 0; v < 8; ++v) {
            const int mm = v + 8 * half;       // 0..15 (local row)
            const int nn = n0 + lrow;          // 0..127
            float x = c[v] + b1[nn];
            Hs[mm * NODE_SIZE + nn] = x / (1.0f + expf(-x));   // silu
        }
    }
    __syncthreads();

    // -------- stage 2: out = Hs @ W2 + b2; wave w -> tiles 3w..3w+2 of 24
#pragma unroll
    for (int t = 0; t < 3; ++t) {
        const int n0 = (wave * 3 + t) * 16;    // 0..368
        v8f c = {};
#pragma unroll
        for (int k0 = 0; k0 < NODE_SIZE; k0 += 4) {
            const int k = k0 + 2 * half;
            v2f a, b;
            a.x = Hs[lrow * NODE_SIZE + k + 0];
            a.y = Hs[lrow * NODE_SIZE + k + 1];
            b.x = W2[(k + 0) * OUT3 + n0 + lrow];
            b.y = W2[(k + 1) * OUT3 + n0 + lrow];
            c = wmma_f32_4(a, b, c);
        }
#pragma unroll
        for (int v = 0; v < 8; ++v) {
            const int mm = m0 + v + 8 * half;
            const int nn = n0 + lrow;
            so[mm * OUT3 + nn] = c[v] + b2[nn];
        }
    }
}

// ---------------------------------------------------------------------------
// Kernel 2: fused edge filter GEMM + cutoff + gating + message scatter-add.
// One block = 16 edges, 256 threads = 8 waves.
// Wave w owns columns cc = 16w + lrow (0..127) of all three gate thirds.
// ---------------------------------------------------------------------------
__global__ void __launch_bounds__(256)
edge_message_kernel(const float* __restrict__ edge_state,  // (E,20)
                    const float* __restrict__ edge_vec,    // (E,3)
                    const float* __restrict__ edge_dist,   // (E,1)
                    const int*   __restrict__ edges,       // (E,2)
                    const float* __restrict__ Wf,          // (20,384)
                    const float* __restrict__ bf,          // (384)
                    const float* __restrict__ so,          // (N,384)
                    const float* __restrict__ ns,          // (N,128)
                    const float* __restrict__ nsv,         // (N,3,128)
                    float* __restrict__ out_s,             // (N,128) accum
                    float* __restrict__ out_v)             // (N,3,128) accum
{
    __shared__ float es_s[16 * EDGE_SIZE];   // edge_state tile
    __shared__ int   src_s[16];
    __shared__ int   dst_s[16];
    __shared__ float cut_s[16];
    __shared__ float env_s[16 * 3];

    const int e0   = blockIdx.x * 16;
    const int tid  = threadIdx.x;
    const int wave = tid >> 5;
    const int lane = tid & 31;
    const int half = lane >> 4;
    const int lrow = lane & 15;

    // stage edge_state tile (contiguous 320 floats)
    for (int i = tid; i < 16 * EDGE_SIZE; i += 256)
        es_s[i] = edge_state[e0 * EDGE_SIZE + i];

    // per-edge scalars: env, cosine cutoff, src/dst
    if (tid < 16) {
        const int e = e0 + tid;
        src_s[tid] = edges[2 * e + 0];
        dst_s[tid] = edges[2 * e + 1];
        const float d = edge_dist[e];
        cut_s[tid] = (d < CUTOFF_R) ? 0.5f * (cosf(PI_F * d / CUTOFF_R) + 1.0f) : 0.0f;
        float vx = edge_vec[3 * e + 0];
        float vy = edge_vec[3 * e + 1];
        float vz = edge_vec[3 * e + 2];
        float nrm = sqrtf(vx * vx + vy * vy + vz * vz);
        float inv = 1.0f / fmaxf(nrm, 1e-12f);
        env_s[3 * tid + 0] = vx * inv;
        env_s[3 * tid + 1] = vy * inv;
        env_s[3 * tid + 2] = vz * inv;
    }
    __syncthreads();

    // GEMM: 16x20 edge_state tile x W_filter columns for the 3 thirds
    v8f acc[3];
#pragma unroll
    for (int third = 0; third < 3; ++third) {
        const int n0 = third * NODE_SIZE + wave * 16;
        v8f c = {};
#pragma unroll
        for (int k0 = 0; k0 < EDGE_SIZE; k0 += 4) {   // 20 = 5 * 4 exactly
            const int k = k0 + 2 * half;
            v2f a, b;
            a.x = es_s[lrow * EDGE_SIZE + k + 0];
            a.y = es_s[lrow * EDGE_SIZE + k + 1];
            b.x = Wf[(k + 0) * OUT3 + n0 + lrow];
            b.y = Wf[(k + 1) * OUT3 + n0 + lrow];
            c = wmma_f32_4(a, b, c);
        }
        acc[third] = c;
    }

    // fused epilogue: bias, cutoff, gate by scalar_output[src], scatter to dst
    const int cc = wave * 16 + lrow;                  // column 0..127
#pragma unroll
    for (int v = 0; v < 8; ++v) {
        const int el = v + 8 * half;                  // local edge 0..15
        const int s  = src_s[el];
        const int dd = dst_s[el];
        const float ct = cut_s[el];

        const float f_gsv = (acc[0][v] + bf[cc])             * ct * so[s * OUT3 + cc];
        const float f_gev = (acc[1][v] + bf[NODE_SIZE + cc]) * ct * so[s * OUT3 + NODE_SIZE + cc];
        const float f_gns = (acc[2][v] + bf[2*NODE_SIZE+cc]) * ct * so[s * OUT3 + 2*NODE_SIZE + cc];

        // scalar message
        atomicAdd(&out_s[dd * NODE_SIZE + cc], ns[s * NODE_SIZE + cc] * f_gns);

        // vector messages (3 spatial dims)
#pragma unroll
        for (int d = 0; d < 3; ++d) {
            const float msg = nsv[s * OUT3 + d * NODE_SIZE + cc] * f_gsv
                            + env_s[3 * el + d] * f_gev;
            atomicAdd(&out_v[dd * OUT3 + d * NODE_SIZE + cc], msg);
        }
    }
}

// ---------------------------------------------------------------------------
extern "C" void kernel_launch(void* const* d_in, const int* in_sizes, int n_in,
                              void* d_out, int out_size, void* d_ws, size_t ws_size,
                              hipStream_t stream) {
    const float* ns   = (const float*)d_in[0];
    const float* nsv  = (const float*)d_in[1];
    const float* es   = (const float*)d_in[2];
    const float* evec = (const float*)d_in[3];
    const float* ed   = (const float*)d_in[4];
    const int*   edg  = (const int*)  d_in[5];
    const float* Wf   = (const float*)d_in[6];
    const float* bf   = (const float*)d_in[7];
    const float* W1   = (const float*)d_in[8];
    const float* b1   = (const float*)d_in[9];
    const float* W2   = (const float*)d_in[10];
    const float* b2   = (const float*)d_in[11];

    float* out_s = (float*)d_out;                       // (N,128)
    float* out_v = out_s + (size_t)NUM_NODES * NODE_SIZE; // (N,3,128)
    float* so    = (float*)d_ws;                        // (N,384) = 30.72 MB scratch

    // residual init: out = node_state (atomics accumulate messages on top)
    hipMemcpyAsync(out_s, ns,  (size_t)NUM_NODES * NODE_SIZE * sizeof(float),
                   hipMemcpyDeviceToDevice, stream);
    hipMemcpyAsync(out_v, nsv, (size_t)NUM_NODES * 3 * NODE_SIZE * sizeof(float),
                   hipMemcpyDeviceToDevice, stream);

    node_mlp_kernel<<<NUM_NODES / 16, 256, 0, stream>>>(ns, W1, b1, W2, b2, so);

    edge_message_kernel<<<NUM_EDGES / 16, 256, 0, stream>>>(
        es, evec, ed, edg, Wf, bf, so, ns, nsv, out_s, out_v);
}